// DirectedMessagePassingLayer_65807488909810
// MI455X (gfx1250) — compile-verified
//
#include <hip/hip_runtime.h>
#include <hip/hip_bf16.h>

#define D_FEAT 128
#define K_TOT  384
#define LN_EPS 1e-5f

typedef __attribute__((ext_vector_type(2))) float v2f;
typedef __attribute__((ext_vector_type(8))) float v8f;

// ---------------------------------------------------------------------------
// Zero the aggregation workspace (float4 granularity).
// ---------------------------------------------------------------------------
__global__ void zero_kernel(float4* __restrict__ p, int n4) {
    int i = blockIdx.x * blockDim.x + threadIdx.x;
    if (i < n4) p[i] = make_float4(0.f, 0.f, 0.f, 0.f);
}

// ---------------------------------------------------------------------------
// COO SpMM scatter: one wave32 per edge, 4 features per lane.
// agg[row] += val * x[col]   via L2-resident global_atomic_add_f32.
// ---------------------------------------------------------------------------
__global__ void spmm_kernel(const int* __restrict__ rows,
                            const int* __restrict__ cols,
                            const float* __restrict__ vals,
                            const float* __restrict__ x,
                            float* __restrict__ agg,
                            int n_edges) {
    const int lane   = threadIdx.x & 31;
    const int wave   = (blockIdx.x * blockDim.x + threadIdx.x) >> 5;
    const int nwaves = (gridDim.x * blockDim.x) >> 5;
    for (int e = wave; e < n_edges; e += nwaves) {
        const int   r = rows[e];
        const int   c = cols[e];
        const float v = vals[e];
        const float4 xv = *(const float4*)(x + (size_t)c * D_FEAT + lane * 4);
        float* dst = agg + (size_t)r * D_FEAT + lane * 4;
        __hip_atomic_fetch_add(dst + 0, v * xv.x, __ATOMIC_RELAXED, __HIP_MEMORY_SCOPE_AGENT);
        __hip_atomic_fetch_add(dst + 1, v * xv.y, __ATOMIC_RELAXED, __HIP_MEMORY_SCOPE_AGENT);
        __hip_atomic_fetch_add(dst + 2, v * xv.z, __ATOMIC_RELAXED, __HIP_MEMORY_SCOPE_AGENT);
        __hip_atomic_fetch_add(dst + 3, v * xv.w, __ATOMIC_RELAXED, __HIP_MEMORY_SCOPE_AGENT);
    }
}

// ---------------------------------------------------------------------------
// Fused GEMM (f32 WMMA 16x16x4) + bias + LayerNorm + ReLU.
// One block = one 16-node tile; 8 waves, wave w owns output columns [16w,16w+16).
//   h[m][n] = sum_k A[m][k] * Wcat[n][k] + b[n],  A = [x | agg_in | agg_out]
// ---------------------------------------------------------------------------
#define A_STRIDE 388   // 384 + 4 pad (banks: 388 % 64 = 4 -> conflict-free cols)
#define H_STRIDE 132   // 128 + 4 pad

__global__ void __launch_bounds__(256)
fused_gemm_ln_relu(const float* __restrict__ x,
                   const float* __restrict__ agg_in,
                   const float* __restrict__ agg_out,
                   const float* __restrict__ Wself,
                   const float* __restrict__ Win,
                   const float* __restrict__ Wout,
                   const float* __restrict__ bias,
                   const float* __restrict__ gamma,
                   const float* __restrict__ beta,
                   float* __restrict__ out) {
    __shared__ float At[16 * A_STRIDE];
    __shared__ float Ht[16 * H_STRIDE];
    __shared__ float mu_s[16];
    __shared__ float rs_s[16];

    const int tid   = threadIdx.x;
    const int node0 = blockIdx.x * 16;

    // Stage A-tile: 16 rows x 96 float4 (x:32, agg_in:32, agg_out:32)
    for (int idx = tid; idx < 16 * 96; idx += 256) {
        const int r = idx / 96;
        const int q = idx % 96;
        const float* src;
        int k4;
        if (q < 32)      { src = x;       k4 = q;      }
        else if (q < 64) { src = agg_in;  k4 = q - 32; }
        else             { src = agg_out; k4 = q - 64; }
        const float4 vv = *(const float4*)(src + (size_t)(node0 + r) * D_FEAT + k4 * 4);
        *(float4*)(&At[r * A_STRIDE + q * 4]) = vv;
    }
    __syncthreads();

    const int lane  = tid & 31;
    const int w     = tid >> 5;       // wave id 0..7
    const int nl    = lane & 15;
    const int khalf = lane >> 4;      // 0 or 1
    const int n     = w * 16 + nl;    // global output column
    const int m     = nl;             // A-fragment row

    // K loop: 3 weight segments x 32 steps of K=4
    v8f acc = {};
    #pragma unroll
    for (int seg = 0; seg < 3; ++seg) {
        const float* Wp = (seg == 0 ? Wself : (seg == 1 ? Win : Wout))
                          + (size_t)n * D_FEAT + 2 * khalf;
        const float* Ap = &At[m * A_STRIDE + seg * D_FEAT + 2 * khalf];
        #pragma unroll 8
        for (int kk = 0; kk < D_FEAT; kk += 4) {
            // A 16x4: VGPR v = A[m][k0 + 2*khalf + v]   (ISA 7.12.2)
            v2f a = *(const v2f*)(Ap + kk);
            // B 4x16 (Wcat^T): mirrored layout, lane n, K in VGPRs
            v2f b = *(const v2f*)(Wp + kk);
            acc = __builtin_amdgcn_wmma_f32_16x16x4_f32(
                      false, a, false, b, (short)0, acc, false, false);
        }
    }

    // Bias + spill h tile to LDS (C/D layout: VGPR r -> M = r + 8*khalf)
    const float bn = bias[n];
    const int   mh = khalf * 8;
    #pragma unroll
    for (int r = 0; r < 8; ++r)
        Ht[(r + mh) * H_STRIDE + n] = acc[r] + bn;
    __syncthreads();

    // Per-row LayerNorm statistics
    if (tid < 16) {
        float s = 0.f, s2 = 0.f;
        for (int c = 0; c < D_FEAT; ++c) {
            const float v = Ht[tid * H_STRIDE + c];
            s  += v;
            s2 += v * v;
        }
        const float mu  = s * (1.f / D_FEAT);
        const float var = s2 * (1.f / D_FEAT) - mu * mu;
        mu_s[tid] = mu;
        rs_s[tid] = rsqrtf(var + LN_EPS);
    }
    __syncthreads();

    // Normalize, affine, ReLU, store (coalesced: 8 floats per thread)
    {
        const int r  = tid >> 4;
        const int c0 = (tid & 15) * 8;
        const float mu = mu_s[r];
        const float rs = rs_s[r];
        float* op = out + (size_t)(node0 + r) * D_FEAT + c0;
        #pragma unroll
        for (int j = 0; j < 8; ++j) {
            float v = (Ht[r * H_STRIDE + c0 + j] - mu) * rs * gamma[c0 + j] + beta[c0 + j];
            op[j] = v > 0.f ? v : 0.f;
        }
    }
}

// ---------------------------------------------------------------------------
extern "C" void kernel_launch(void* const* d_in, const int* in_sizes, int n_in,
                              void* d_out, int out_size, void* d_ws, size_t ws_size,
                              hipStream_t stream) {
    const float* x      = (const float*)d_in[0];
    const int*   ai_r   = (const int*)  d_in[1];
    const int*   ai_c   = (const int*)  d_in[2];
    const float* ai_v   = (const float*)d_in[3];
    const int*   ao_r   = (const int*)  d_in[4];
    const int*   ao_c   = (const int*)  d_in[5];
    const float* ao_v   = (const float*)d_in[6];
    const float* Wself  = (const float*)d_in[7];
    const float* b_self = (const float*)d_in[8];
    const float* Win    = (const float*)d_in[9];
    const float* Wout   = (const float*)d_in[10];
    const float* gamma  = (const float*)d_in[11];
    const float* beta   = (const float*)d_in[12];
    float*       out    = (float*)d_out;

    const int n_nodes = in_sizes[0] / D_FEAT;   // 50000
    const int n_edges = in_sizes[1];            // 600000

    float* agg_in  = (float*)d_ws;
    float* agg_out = agg_in + (size_t)n_nodes * D_FEAT;

    // 1) zero accumulators
    {
        const int n4 = (2 * n_nodes * D_FEAT) / 4;
        zero_kernel<<<(n4 + 255) / 256, 256, 0, stream>>>((float4*)agg_in, n4);
    }

    // 2) two SpMM scatter passes (one wave per edge, grid-strided)
    {
        const int waves_per_block = 256 / 32;
        int blocks = (n_edges + waves_per_block - 1) / waves_per_block;
        if (blocks > 65535 * 8) blocks = 65535 * 8;
        spmm_kernel<<<blocks, 256, 0, stream>>>(ai_r, ai_c, ai_v, x, agg_in,  n_edges);
        spmm_kernel<<<blocks, 256, 0, stream>>>(ao_r, ao_c, ao_v, x, agg_out, n_edges);
    }

    // 3) fused WMMA GEMM + LayerNorm + ReLU  (50000 nodes = 3125 tiles of 16)
    {
        const int tiles = n_nodes / 16;
        fused_gemm_ln_relu<<<tiles, 256, 0, stream>>>(
            x, agg_in, agg_out, Wself, Win, Wout, b_self, gamma, beta, out);
    }
}